// AngleFreqEnhance_24962349924395
// MI455X (gfx1250) — compile-verified
//
#include <hip/hip_runtime.h>
#include <hip/hip_bf16.h>
#include <math.h>

// ---------------- problem constants ----------------
#define B_    8
#define C_    256
#define CM_   16
#define H_    128
#define W_    128
#define HW_   16384              // H_*W_
#define NANG  8
#define NRAD  9                  // 64/8 + 1
#define EPS_  1e-8f
#define CLIP_ 10.0f
#define PI_F  3.14159265358979323846f

typedef __attribute__((ext_vector_type(2))) float v2f;
typedef __attribute__((ext_vector_type(8))) float v8f;

// ---------------- workspace layout (bytes) ----------------
// [0, 1MB)      : gain_s  (CM * H * W floats, stored at UNshifted freq coords)
// [1MB, 9MB)    : x_proj  (B * CM * HW floats)  -- later reused as x_enh
// [9MB, 25MB)   : freq    (B*CM planes * HW float2)
#define GAIN_OFF   (0)
#define XPROJ_OFF  (1u << 20)
#define FREQ_OFF   (9u << 20)

// =====================================================================
// Kernel 1: gain precompute at pre-ifftshifted coordinates.
// =====================================================================
__global__ void afe_gain_kernel(const float* __restrict__ wraw,
                                float* __restrict__ gain_s) {
    int idx = blockIdx.x * blockDim.x + threadIdx.x;   // CM*HW threads
    int m  = idx >> 14;
    int ky = (idx >> 7) & 127;
    int kx = idx & 127;
    int h = (ky + 64) & 127;
    int w = (kx + 64) & 127;
    float dy = (float)h - 64.0f;
    float dx = (float)w - 64.0f;
    float r = sqrtf(dy * dy + dx * dx);
    float theta = fmodf(atan2f(dy, dx) + PI_F, PI_F);
    int ridx = (int)floorf(r * 0.125f);                 // r / RADIUS_WIDTH
    ridx = min(max(ridx, 0), NRAD - 1);

    const float delta = PI_F / (float)NANG;
    const float hwid  = 1.5f * delta * 0.5f;            // OVERLAP*delta/2
    float aw[NANG];
    float den = EPS_;
#pragma unroll
    for (int a = 0; a < NANG; ++a) {
        float center = ((float)a + 0.5f) * delta;
        float dist = fabsf(theta - center);
        float v = (dist < hwid) ? (1.0f - dist / hwid) : 0.0f;
        v = fmaxf(v, 0.0f);
        aw[a] = v;
        den += v;
    }
    float g = 0.0f;
#pragma unroll
    for (int a = 0; a < NANG; ++a) {
        float wt = 1.0f + tanhf(wraw[(m * NANG + a) * NRAD + ridx]);
        g += (aw[a] / den) * wt;
    }
    bool valid = (r >= 0.5f) && (r > 0.3f * 64.0f);     // HF_RATIO*max(cy,cx)
    gain_s[idx] = valid ? g : 1.0f;
}

// =====================================================================
// Kernel 2: x_proj[b,m,n] = sum_c w_in[m,c] * x[b,c,n]
// One wave per 16x16 (m x n) tile; V_WMMA_F32_16X16X4_F32, 64 k-steps.
// =====================================================================
__global__ void afe_proj_in_kernel(const float* __restrict__ x,
                                   const float* __restrict__ w_in,
                                   float* __restrict__ xproj) {
    int lane = threadIdx.x & 31;
    int gw = blockIdx.x * 8 + (threadIdx.x >> 5);       // 8192 waves total
    int b  = gw >> 10;                                  // / (HW/16)
    int n0 = (gw & 1023) << 4;

    int row = lane & 15;
    int kb  = (lane >> 4) << 1;                         // 0 or 2

    const float* xb   = x + (size_t)b * C_ * HW_;
    const float* wrow = w_in + (size_t)row * C_;        // w_in[m=row][c]

    v8f acc = {};
#pragma unroll 4
    for (int k = 0; k < C_; k += 4) {
        v2f a, bb;
        a.x  = wrow[k + kb];
        a.y  = wrow[k + kb + 1];
        bb.x = xb[(size_t)(k + kb) * HW_ + n0 + row];
        bb.y = xb[(size_t)(k + kb + 1) * HW_ + n0 + row];
        acc = __builtin_amdgcn_wmma_f32_16x16x4_f32(
            false, a, false, bb, (short)0, acc, false, false);
    }
    float* xp = xproj + (size_t)b * CM_ * HW_;
#pragma unroll
    for (int v = 0; v < 8; ++v) {
        int mrow = v + ((lane >> 4) << 3);
        xp[(size_t)mrow * HW_ + n0 + (lane & 15)] = acc[v];
    }
}

// =====================================================================
// 128-point radix-2 FFT helpers (64 worker lanes per line, data in LDS)
// =====================================================================
__device__ __forceinline__ unsigned brev7(unsigned x) { return __brev(x) >> 25; }

__device__ __forceinline__ void fft128_stages(float2* ln, unsigned lane, float sgn) {
#pragma unroll
    for (int s = 1; s <= 7; ++s) {
        unsigned m = 1u << s, half = m >> 1;
        unsigned lo = lane & (half - 1);
        unsigned j  = ((lane >> (s - 1)) << s) + lo;
        float ang = sgn * 2.0f * PI_F * (float)lo / (float)m;  // sgn=-1 fwd, +1 inv
        float wr, wi;
        __sincosf(ang, &wi, &wr);
        __syncthreads();
        float2 a = ln[j];
        float2 b = ln[j + half];
        float tr = b.x * wr - b.y * wi;
        float ti = b.x * wi + b.y * wr;
        ln[j]        = make_float2(a.x + tr, a.y + ti);
        ln[j + half] = make_float2(a.x - tr, a.y - ti);
    }
}

__device__ __forceinline__ void bitrev_permute(float2* ln, unsigned lane) {
#pragma unroll
    for (int rep = 0; rep < 2; ++rep) {
        unsigned j  = lane + rep * 64u;
        unsigned rj = brev7(j);
        if (j < rj) { float2 t = ln[j]; ln[j] = ln[rj]; ln[rj] = t; }
    }
}

// gfx1250 async LDS<->global helpers (inline asm; ASYNCcnt-tracked).
__device__ __forceinline__ void async_load_b64_to_lds(unsigned lds_byte_off,
                                                      const void* gaddr) {
    asm volatile("global_load_async_to_lds_b64 %0, %1, off"
                 :: "v"(lds_byte_off), "v"(gaddr) : "memory");
}
__device__ __forceinline__ void async_store_b64_from_lds(void* gaddr,
                                                         unsigned lds_byte_off) {
    asm volatile("global_store_async_from_lds_b64 %0, %1, off"
                 :: "v"(gaddr), "v"(lds_byte_off) : "memory");
}
__device__ __forceinline__ void wait_asynccnt0() {
    asm volatile("s_wait_asynccnt 0" ::: "memory");
}

// =====================================================================
// Kernel 3: forward FFT along rows (real input -> complex).
// 4 lines per 256-thread block; line = plane*128 + h.
// Output written via async store straight from LDS.
// =====================================================================
__global__ void afe_fft_rows_fwd(const float* __restrict__ xp,
                                 float2* __restrict__ freq) {
    extern __shared__ float2 smem[];            // 4 * 128 float2 (dyn, base 0)
    int tid = threadIdx.x;
    int lgrp = tid >> 6;
    unsigned lane = tid & 63;
    int line = blockIdx.x * 4 + lgrp;
    const float* src = xp + (size_t)line * 128;
    float2* ln = smem + lgrp * 128;
    ln[brev7(lane)]       = make_float2(src[lane], 0.0f);
    ln[brev7(lane + 64u)] = make_float2(src[lane + 64], 0.0f);
    fft128_stages(ln, lane, -1.0f);
    __syncthreads();
    float2* dst = freq + (size_t)line * 128;
    unsigned lbase = (unsigned)(lgrp * 128) * 8u;
    async_store_b64_from_lds((void*)(dst + lane),      lbase + lane * 8u);
    async_store_b64_from_lds((void*)(dst + lane + 64), lbase + (lane + 64u) * 8u);
    // drained by implicit wait-idle at s_endpgm
}

// =====================================================================
// Kernel 4: columns: forward FFT, ortho scale (1/128), gain+eps pointwise,
// inverse FFT. In-place on freq. task = plane*128 + kx.
// Column gathers/scatters (stride 1024B per element) go via async LDS path,
// with the bit-reversal folded into the async gather LDS addresses.
// =====================================================================
__global__ void afe_fft_cols_gain(float2* __restrict__ freq,
                                  const float* __restrict__ gain_s) {
    extern __shared__ float2 smem[];
    int tid = threadIdx.x;
    int lgrp = tid >> 6;
    unsigned lane = tid & 63;
    int task  = blockIdx.x * 4 + lgrp;
    int plane = task >> 7;
    int kx    = task & 127;
    int mch   = plane & (CM_ - 1);
    float2* base = freq + (size_t)plane * HW_ + kx;
    float2* ln = smem + lgrp * 128;
    unsigned lbase = (unsigned)(lgrp * 128) * 8u;

    async_load_b64_to_lds(lbase + brev7(lane) * 8u,
                          (const void*)(base + (size_t)lane * 128));
    async_load_b64_to_lds(lbase + brev7(lane + 64u) * 8u,
                          (const void*)(base + (size_t)(lane + 64) * 128));
    wait_asynccnt0();

    fft128_stages(ln, lane, -1.0f);
    __syncthreads();
#pragma unroll
    for (int rep = 0; rep < 2; ++rep) {
        int ky = (int)lane + rep * 64;
        float2 v = ln[ky];
        v.x *= (1.0f / 128.0f);                 // ortho scale for forward 2D
        v.y *= (1.0f / 128.0f);
        float g = gain_s[((size_t)mch << 14) + ((size_t)ky << 7) + kx];
        float mag = sqrtf(v.x * v.x + v.y * v.y);
        if (mag > 0.0f) {
            float s = g * (mag + EPS_) / mag;   // (|f|+eps)*gain*e^{i angle}
            v.x *= s; v.y *= s;
        } else {
            v.x = g * EPS_; v.y = 0.0f;         // angle(0)=0 -> e^{i0}=1
        }
        ln[ky] = v;
    }
    __syncthreads();
    bitrev_permute(ln, lane);
    fft128_stages(ln, lane, +1.0f);             // inverse (unscaled)
    __syncthreads();
    async_store_b64_from_lds((void*)(base + (size_t)lane * 128),
                             lbase + lane * 8u);
    async_store_b64_from_lds((void*)(base + (size_t)(lane + 64) * 128),
                             lbase + (lane + 64u) * 8u);
}

// =====================================================================
// Kernel 5: inverse FFT along rows (async load into bitrev LDS slots),
// apply 1/128 (ortho inverse total), take real part -> x_enh.
// =====================================================================
__global__ void afe_fft_rows_inv(const float2* __restrict__ freq,
                                 float* __restrict__ xe) {
    extern __shared__ float2 smem[];
    int tid = threadIdx.x;
    int lgrp = tid >> 6;
    unsigned lane = tid & 63;
    int line = blockIdx.x * 4 + lgrp;
    const float2* src = freq + (size_t)line * 128;
    float2* ln = smem + lgrp * 128;
    unsigned lbase = (unsigned)(lgrp * 128) * 8u;

    async_load_b64_to_lds(lbase + brev7(lane) * 8u,       (const void*)(src + lane));
    async_load_b64_to_lds(lbase + brev7(lane + 64u) * 8u, (const void*)(src + lane + 64));
    wait_asynccnt0();

    fft128_stages(ln, lane, +1.0f);
    __syncthreads();
    float* dst = xe + (size_t)line * 128;
    dst[lane]      = ln[lane].x * (1.0f / 128.0f);
    dst[lane + 64] = ln[lane + 64].x * (1.0f / 128.0f);
}

// =====================================================================
// Kernel 6: out[b,c,n] = x[b,c,n] + clip(sum_m w_out[c,m]*x_enh[b,m,n])
// One wave per 16x16 (c x n) tile; WMMA f32 16x16x4, K=16 (4 steps).
// =====================================================================
__global__ void afe_proj_out_kernel(const float* __restrict__ x,
                                    const float* __restrict__ w_out,
                                    const float* __restrict__ xe,
                                    float* __restrict__ out) {
    int lane = threadIdx.x & 31;
    int gw = blockIdx.x * 8 + (threadIdx.x >> 5);       // 131072 waves
    int b   = gw >> 14;                                 // / (16 ctiles * 1024 ntiles)
    int rem = gw & 16383;
    int c0  = (rem >> 10) << 4;
    int n0  = (rem & 1023) << 4;

    int row = lane & 15;
    int kb  = (lane >> 4) << 1;

    const float* xeb  = xe + (size_t)b * CM_ * HW_;
    const float* wrow = w_out + (size_t)(c0 + row) * CM_;  // w_out[c][m]

    v8f acc = {};
#pragma unroll
    for (int k = 0; k < CM_; k += 4) {
        v2f a, bb;
        a.x  = wrow[k + kb];
        a.y  = wrow[k + kb + 1];
        bb.x = xeb[(size_t)(k + kb) * HW_ + n0 + row];
        bb.y = xeb[(size_t)(k + kb + 1) * HW_ + n0 + row];
        acc = __builtin_amdgcn_wmma_f32_16x16x4_f32(
            false, a, false, bb, (short)0, acc, false, false);
    }

    const float* xb = x   + ((size_t)b * C_ + c0) * HW_;
    float*       ob = out + ((size_t)b * C_ + c0) * HW_;
#pragma unroll
    for (int v = 0; v < 8; ++v) {
        int crow = v + ((lane >> 4) << 3);
        size_t idx = (size_t)crow * HW_ + n0 + (lane & 15);
        float val = fminf(fmaxf(acc[v], -CLIP_), CLIP_);
        ob[idx] = xb[idx] + val;
    }
}

// =====================================================================
// launch
// =====================================================================
extern "C" void kernel_launch(void* const* d_in, const int* in_sizes, int n_in,
                              void* d_out, int out_size, void* d_ws, size_t ws_size,
                              hipStream_t stream) {
    const float* x     = (const float*)d_in[0];
    const float* w_in  = (const float*)d_in[1];
    const float* w_out = (const float*)d_in[2];
    const float* wraw  = (const float*)d_in[3];
    float* out = (float*)d_out;

    char* ws = (char*)d_ws;                 // needs 25 MiB
    float*  gain_s = (float*)(ws + GAIN_OFF);
    float*  xproj  = (float*)(ws + XPROJ_OFF);   // reused as x_enh
    float2* freq   = (float2*)(ws + FREQ_OFF);

    const size_t fft_lds = 4 * 128 * sizeof(float2);   // 4 KiB dynamic LDS

    // 1. gain:   CM*HW threads
    afe_gain_kernel<<<(CM_ * HW_) / 256, 256, 0, stream>>>(wraw, gain_s);
    // 2. proj in: 8192 waves -> 1024 blocks of 8 waves
    afe_proj_in_kernel<<<1024, 256, 0, stream>>>(x, w_in, xproj);
    // 3-5. FFT passes: 16384 lines each, 4 lines/block
    afe_fft_rows_fwd<<<4096, 256, fft_lds, stream>>>(xproj, freq);
    afe_fft_cols_gain<<<4096, 256, fft_lds, stream>>>(freq, gain_s);
    afe_fft_rows_inv<<<4096, 256, fft_lds, stream>>>(freq, xproj);
    // 6. proj out + clip + residual: 131072 waves -> 16384 blocks
    afe_proj_out_kernel<<<16384, 256, 0, stream>>>(x, w_out, xproj, out);
}